// TCTRN_Attention_30219389894947
// MI455X (gfx1250) — compile-verified
//
#include <hip/hip_runtime.h>
#include <hip/hip_bf16.h>
#include <float.h>

#define B_      64
#define T_      1000
#define RNN_    1024
#define EMB_    512
#define ATT_    128
#define LOC_    32
#define KS_     31
#define PAD_    15
#define TT_     64              // t-values per block in the fused kernel
#define NTILE_  16              // ceil(1000/64)
#define NCHUNK_ 4
#define CHUNK_  250             // t per chunk (2 interleaved halves of 125)
#define NPART_  (NCHUNK_ * 2)   // deterministic partials per (b,e)

typedef __attribute__((ext_vector_type(2))) float v2f;
typedef __attribute__((ext_vector_type(8))) float v8f;

// ---------------------------------------------------------------------------
// Kernel 0: pq[b][a] = sum_k hidden[b][k] * W_query[a][k]   (64x128, K=1024)
// ---------------------------------------------------------------------------
__global__ __launch_bounds__(256)
void k_pq(const float* __restrict__ h, const float* __restrict__ wq,
          float* __restrict__ pq)
{
    int idx = blockIdx.x * 256 + threadIdx.x;   // 8192 outputs
    int b = idx >> 7;
    int a = idx & (ATT_ - 1);
    const float4* h4 = (const float4*)(h + (size_t)b * RNN_);
    const float4* w4 = (const float4*)(wq + (size_t)a * RNN_);
    float acc = 0.f;
    for (int k = 0; k < RNN_ / 4; ++k) {
        float4 x = h4[k], y = w4[k];
        acc = fmaf(x.x, y.x, acc);
        acc = fmaf(x.y, y.y, acc);
        acc = fmaf(x.z, y.z, acc);
        acc = fmaf(x.w, y.w, acc);
    }
    pq[idx] = acc;
}

// ---------------------------------------------------------------------------
// Kernel 1 (WMMA): fused conv1d -> W_loc projection -> tanh energy
//   grid = (16 t-tiles, 64 batches), 256 threads = 8 wave32
//   D(64t x 128a) = loc(64t x 32c) @ W_loc^T(32c x 128a) via v_wmma_f32_16x16x4
// ---------------------------------------------------------------------------
__global__ __launch_bounds__(256)
void k_conv_proj_energy(const float* __restrict__ awc,    // (B,2,T)
                        const float* __restrict__ convw,  // (32,2,31)
                        const float* __restrict__ wloc,   // (128,32)
                        const float* __restrict__ pq,     // (B,128)
                        const float* __restrict__ pm,     // (B,T,128)
                        const float* __restrict__ vvec,   // (128)
                        float* __restrict__ energy)       // (B,T)
{
    const int tile = blockIdx.x;
    const int b    = blockIdx.y;
    const int t0   = tile * TT_;
    const int tid  = threadIdx.x;
    const int lane = tid & 31;
    const int wave = tid >> 5;

    __shared__ float x_s[2][TT_ + KS_ - 1];     // conv input halo (94 per ch)
    __shared__ float loc_s[TT_][LOC_ + 1];      // conv output, pad->33 (bank-safe)
    __shared__ float wl_s[ATT_][LOC_ + 1];      // W_loc staged, padded
    __shared__ float pq_s[ATT_];
    __shared__ float v_s[ATT_];
    __shared__ float en_s[TT_];

    // ---- stage shared data -------------------------------------------------
    for (int i = tid; i < ATT_ * LOC_; i += 256)
        wl_s[i >> 5][i & 31] = wloc[i];
    if (tid < ATT_) {
        pq_s[tid] = pq[b * ATT_ + tid];
        v_s[tid]  = vvec[tid];
    }
    for (int i = tid; i < 2 * (TT_ + KS_ - 1); i += 256) {
        int ch = i / (TT_ + KS_ - 1);
        int j  = i % (TT_ + KS_ - 1);
        int t  = t0 + j - PAD_;                 // SAME padding
        x_s[ch][j] = (t >= 0 && t < T_) ? awc[((size_t)b * 2 + ch) * T_ + t] : 0.f;
    }
    if (tid < TT_) en_s[tid] = 0.f;
    __syncthreads();

    // ---- conv1d: 32 channels x 64 t, 62 MACs each --------------------------
    for (int r = 0; r < (LOC_ * TT_) / 256; ++r) {
        int idx = tid + r * 256;
        int c = idx >> 6;                       // channel 0..31
        int t = idx & 63;                       // local t 0..63
        float acc = 0.f;
        #pragma unroll
        for (int ch = 0; ch < 2; ++ch) {
            const float* wk = convw + (c * 2 + ch) * KS_;
            #pragma unroll
            for (int k = 0; k < KS_; ++k)
                acc = fmaf(x_s[ch][t + k], wk[k], acc);   // x(t0+t+k-15)
        }
        loc_s[t][c] = acc;
    }
    __syncthreads();

    // ---- WMMA projection + tanh energy epilogue ----------------------------
    // wave owns ATT subtile [wave*16, wave*16+16); loops over 4 t-subtiles.
    const int an = wave * 16 + (lane & 15);     // N index (ATT dim)
    const int hi = lane >> 4;                   // lane half (K split / M split)
    const int m  = lane & 15;                   // A-matrix row within subtile

    for (int tsub = 0; tsub < 4; ++tsub) {
        const int tb = tsub * 16;
        v8f acc = {0.f, 0.f, 0.f, 0.f, 0.f, 0.f, 0.f, 0.f};
        #pragma unroll
        for (int k0 = 0; k0 < LOC_; k0 += 4) {
            const int ka = k0 + 2 * hi;         // ISA A/B VGPR layout: K0/K2, K1/K3
            v2f Am, Bm;
            Am.x = loc_s[tb + m][ka];
            Am.y = loc_s[tb + m][ka + 1];
            Bm.x = wl_s[an][ka];                // B[k][n] = W_loc[n][k]
            Bm.y = wl_s[an][ka + 1];
            acc = __builtin_amdgcn_wmma_f32_16x16x4_f32(
                false, Am, false, Bm, (short)0, acc, false, false);
        }
        // D layout: VGPR j -> rows tb+j (lanes 0-15) and tb+j+8 (lanes 16-31)
        #pragma unroll
        for (int j = 0; j < 8; ++j) {
            const int t_loc = tb + j + 8 * hi;
            int tg = t0 + t_loc;
            int tc = tg < T_ ? tg : T_ - 1;     // clamp tail tile (no OOB reads)
            float val = acc[j] + pq_s[an]
                      + pm[((size_t)b * T_ + tc) * ATT_ + an];
            float e = tanhf(val) * v_s[an];
            // reduce the 16-lane N dimension (stays within each lane half)
            e += __shfl_xor(e, 1, 32);
            e += __shfl_xor(e, 2, 32);
            e += __shfl_xor(e, 4, 32);
            e += __shfl_xor(e, 8, 32);
            if ((lane & 15) == 0) atomicAdd(&en_s[t_loc], e);  // ds_add_f32
        }
    }
    __syncthreads();
    if (tid < TT_) {
        int tg = t0 + tid;
        if (tg < T_) energy[b * T_ + tg] = en_s[tid];
    }
}

// ---------------------------------------------------------------------------
// Kernel 2: masked softmax over T per batch; writes attention weights to d_out
// ---------------------------------------------------------------------------
__global__ __launch_bounds__(256)
void k_softmax(const float* __restrict__ energy,
               const unsigned char* __restrict__ mask,
               float* __restrict__ wout)
{
    int b = blockIdx.x;
    int tid = threadIdx.x;
    __shared__ float red[256];
    __shared__ float stat[2];

    float mx = -FLT_MAX;
    for (int t = tid; t < T_; t += 256) {
        float e = mask[b * T_ + t] ? -FLT_MAX : energy[b * T_ + t];
        mx = fmaxf(mx, e);
    }
    red[tid] = mx; __syncthreads();
    for (int s = 128; s > 0; s >>= 1) {
        if (tid < s) red[tid] = fmaxf(red[tid], red[tid + s]);
        __syncthreads();
    }
    if (tid == 0) stat[0] = red[0];
    __syncthreads();
    float bmax = stat[0];

    float sum = 0.f;
    for (int t = tid; t < T_; t += 256) {
        float e  = mask[b * T_ + t] ? -FLT_MAX : energy[b * T_ + t];
        float ex = __expf(e - bmax);
        wout[b * T_ + t] = ex;
        sum += ex;
    }
    red[tid] = sum; __syncthreads();
    for (int s = 128; s > 0; s >>= 1) {
        if (tid < s) red[tid] += red[tid + s];
        __syncthreads();
    }
    if (tid == 0) stat[1] = red[0];
    __syncthreads();
    float inv = 1.0f / stat[1];
    for (int t = tid; t < T_; t += 256)
        wout[b * T_ + t] *= inv;
}

// ---------------------------------------------------------------------------
// Kernel 3a: partial context over t-chunks — the 131 MB stream.
//   global_load_b128: each thread owns one float4 column (e4 = tid&127); the
//   two 128-thread halves interleave over t (stride 2) -> 512 B per wave per
//   load instruction, 8 deterministic partials per (b, e4).
//   grid = (4 chunks, 64 batches)
// ---------------------------------------------------------------------------
__global__ __launch_bounds__(256)
void k_ctx_partial(const float* __restrict__ w,     // (B,T) attention weights
                   const float* __restrict__ mem,   // (B,T,512)
                   float4* __restrict__ part)       // (B,4,2,128) float4
{
    int c = blockIdx.x, b = blockIdx.y;
    int tid = threadIdx.x;
    int g   = tid >> 7;                 // half: t parity
    int e4  = tid & 127;                // float4 column
    int t_begin = c * CHUNK_;

    __shared__ float w_s[CHUNK_];
    for (int t = tid; t < CHUNK_; t += 256)
        w_s[t] = w[b * T_ + t_begin + t];
    __syncthreads();

    const float4* mem4 = (const float4*)(mem + (size_t)b * T_ * EMB_)
                       + (size_t)(t_begin + g) * (EMB_ / 4) + e4;
    float4 acc = {0.f, 0.f, 0.f, 0.f};
    #pragma unroll 4
    for (int t = g; t < CHUNK_; t += 2) {
        float wt = w_s[t];
        float4 x = *mem4;
        mem4 += 2 * (EMB_ / 4);
        acc.x = fmaf(wt, x.x, acc.x);
        acc.y = fmaf(wt, x.y, acc.y);
        acc.z = fmaf(wt, x.z, acc.z);
        acc.w = fmaf(wt, x.w, acc.w);
    }
    part[(((size_t)b * NCHUNK_ + c) * 2 + g) * (EMB_ / 4) + e4] = acc;
}

// ---------------------------------------------------------------------------
// Kernel 3b: fold the 8 deterministic partials into d_out context (b128 I/O)
// ---------------------------------------------------------------------------
__global__ __launch_bounds__(256)
void k_ctx_reduce(const float4* __restrict__ part, float4* __restrict__ out)
{
    int idx = blockIdx.x * 256 + threadIdx.x;   // 64*128 float4 outputs
    int b  = idx >> 7;
    int e4 = idx & 127;
    const float4* p = part + (size_t)b * NPART_ * (EMB_ / 4) + e4;
    float4 acc = {0.f, 0.f, 0.f, 0.f};
    #pragma unroll
    for (int i = 0; i < NPART_; ++i) {
        float4 x = p[(size_t)i * (EMB_ / 4)];
        acc.x += x.x; acc.y += x.y; acc.z += x.z; acc.w += x.w;
    }
    out[idx] = acc;
}

// ---------------------------------------------------------------------------
extern "C" void kernel_launch(void* const* d_in, const int* in_sizes, int n_in,
                              void* d_out, int out_size, void* d_ws, size_t ws_size,
                              hipStream_t stream)
{
    const float* h    = (const float*)d_in[0];   // attention_hidden_state (B,1024)
    const float* mem  = (const float*)d_in[1];   // memory (B,T,512)
    const float* pm   = (const float*)d_in[2];   // processed_memory (B,T,128)
    const float* awc  = (const float*)d_in[3];   // attention_weights_cat (B,2,T)
    const unsigned char* mask = (const unsigned char*)d_in[4]; // (B,T) bool
    const float* wq   = (const float*)d_in[5];   // W_query (128,1024)
    const float* cw   = (const float*)d_in[6];   // conv_w (32,2,31)
    const float* wl   = (const float*)d_in[7];   // W_loc (128,32)
    const float* vv   = (const float*)d_in[8];   // v (1,128)

    float* ctx_out = (float*)d_out;              // (B,512) first
    float* w_out   = ctx_out + B_ * EMB_;        // (B,T) second

    float* pq     = (float*)d_ws;                // B*128
    float* energy = pq + B_ * ATT_;              // B*T
    float* part   = energy + B_ * T_;            // B*8*512 (16B-aligned: offsets
                                                 // 8192 + 64000 floats -> 16B ok? 
                                                 // 64000*4=256000, 256000%16==0; pq
                                                 // offset 0; energy offset 32768. OK.)

    k_pq<<<(B_ * ATT_) / 256, 256, 0, stream>>>(h, wq, pq);
    k_conv_proj_energy<<<dim3(NTILE_, B_), 256, 0, stream>>>(
        awc, cw, wl, pq, pm, vv, energy);
    k_softmax<<<B_, 256, 0, stream>>>(energy, mask, w_out);
    k_ctx_partial<<<dim3(NCHUNK_, B_), 256, 0, stream>>>(
        w_out, mem, (float4*)part);
    k_ctx_reduce<<<(B_ * EMB_ / 4) / 256, 256, 0, stream>>>(
        (const float4*)part, (float4*)ctx_out);
}